// MetricConv_53807350284472
// MI455X (gfx1250) — compile-verified
//
#include <hip/hip_runtime.h>
#include <hip/hip_bf16.h>

typedef __attribute__((ext_vector_type(2))) float v2f;
typedef __attribute__((ext_vector_type(8))) float v8f;

#define NV   50000
#define NC   128
#define NFO  128
#define NH   32
#define NE   800000
#define EPSV 1e-8f
#define MBLK 96

// ---------------------------------------------------------------------------
// Kernel 1: per-vertex metric  G = (relu(X W1 + b1) W2 + b2) as M, G = M^T M
// Features staged through LDS (pitch 129 -> bank-conflict-free row reads).
// ---------------------------------------------------------------------------
__global__ __launch_bounds__(MBLK) void metric_kernel(
    const float* __restrict__ feat,
    const float* __restrict__ W1, const float* __restrict__ b1,
    const float* __restrict__ W2, const float* __restrict__ b2,
    float* __restrict__ Gout) {
  __shared__ float tile[MBLK][NC + 1];
  const int vbase = blockIdx.x * MBLK;
  const int tid = threadIdx.x;

  for (int idx = tid; idx < MBLK * NC; idx += MBLK) {
    const int v = idx >> 7;        // NC == 128
    const int c = idx & (NC - 1);
    const int vv = vbase + v;
    tile[v][c] = (vv < NV) ? feat[(size_t)vv * NC + c] : 0.0f;
  }
  __syncthreads();

  const int vert = vbase + tid;
  if (vert >= NV) return;

  float h[NH];
#pragma unroll
  for (int j = 0; j < NH; ++j) h[j] = b1[j];
  for (int c = 0; c < NC; ++c) {
    const float xc = tile[tid][c];
#pragma unroll
    for (int j = 0; j < NH; ++j) h[j] = fmaf(xc, W1[c * NH + j], h[j]);
  }
#pragma unroll
  for (int j = 0; j < NH; ++j) h[j] = fmaxf(h[j], 0.0f);

  float m[9];
#pragma unroll
  for (int p = 0; p < 9; ++p) m[p] = b2[p];
  for (int j = 0; j < NH; ++j) {
    const float hj = h[j];
#pragma unroll
    for (int p = 0; p < 9; ++p) m[p] = fmaf(hj, W2[j * 9 + p], m[p]);
  }

  float* g = Gout + (size_t)vert * 9;
#pragma unroll
  for (int i = 0; i < 3; ++i)
#pragma unroll
    for (int j = 0; j < 3; ++j)
      g[i * 3 + j] = m[0 * 3 + i] * m[0 * 3 + j]
                   + m[1 * 3 + i] * m[1 * 3 + j]
                   + m[2 * 3 + i] * m[2 * 3 + j];
}

// ---------------------------------------------------------------------------
// Kernel 2: XW = X @ weights via V_WMMA_F32_16X16X4_F32 (exact fp32 path).
// Block = 8 waves; wave w computes the 16x16 tile (rows blockIdx.x*16, cols w*16).
// A 16x4 f32 layout: lane&15 = M, VGPR0/1 hold K = 2*(lane>>4), +1.
// B 4x16  f32 layout: lane&15 = N, VGPR r holds K = r + 2*(lane>>4).
// C/D: VGPR r -> (M = r + 8*(lane>>4), N = lane&15).
// ---------------------------------------------------------------------------
__global__ __launch_bounds__(256) void xw_wmma(
    const float* __restrict__ X,
    const float* __restrict__ W,
    float* __restrict__ XW) {
  const int lane = threadIdx.x & 31;
  const int wave = threadIdx.x >> 5;
  const int row0 = blockIdx.x * 16;          // NV % 16 == 0 (3125 tiles)
  const int col0 = wave * 16;                // 8 waves cover NFO = 128
  const int m  = lane & 15;
  const int kh = (lane >> 4) * 2;

  v8f acc = {0.f, 0.f, 0.f, 0.f, 0.f, 0.f, 0.f, 0.f};
  const float* ap = X + (size_t)(row0 + m) * NC + kh;   // A row m, k = kh, kh+1
  const float* bp = W + (size_t)kh * NFO + col0 + m;    // B col m, k = kh, kh+1

  for (int kb = 0; kb < NC; kb += 4) {
    v2f a, b;
    a.x = ap[kb];
    a.y = ap[kb + 1];
    b.x = bp[(size_t)kb * NFO];
    b.y = bp[(size_t)(kb + 1) * NFO];
    acc = __builtin_amdgcn_wmma_f32_16x16x4_f32(
        /*neg_a=*/false, a, /*neg_b=*/false, b,
        /*c_mod=*/(short)0, acc, /*reuse_a=*/false, /*reuse_b=*/false);
  }

  const int ccol  = col0 + m;
  const int rbase = row0 + (lane >> 4) * 8;
#pragma unroll
  for (int r = 0; r < 8; ++r)
    XW[(size_t)(rbase + r) * NFO + ccol] = acc[r];
}

// ---------------------------------------------------------------------------
// Kernel 3: per-edge weight w = exp(-t^T * 0.5*(G_s + G_d) * t), rowsum atomics.
// ---------------------------------------------------------------------------
__global__ __launch_bounds__(256) void edge_kernel(
    const float* __restrict__ vtx,
    const int* __restrict__ edges,
    const float* __restrict__ G,
    float* __restrict__ wE,
    float* __restrict__ rowsum) {
  const int e = blockIdx.x * blockDim.x + threadIdx.x;
  if (e >= NE) return;
  const int s = edges[e];
  const int d = edges[NE + e];

  float g[9];
#pragma unroll
  for (int k = 0; k < 9; ++k)
    g[k] = 0.5f * (G[(size_t)s * 9 + k] + G[(size_t)d * 9 + k]);

  const float t0 = vtx[(size_t)d * 3 + 0] - vtx[(size_t)s * 3 + 0];
  const float t1 = vtx[(size_t)d * 3 + 1] - vtx[(size_t)s * 3 + 1];
  const float t2 = vtx[(size_t)d * 3 + 2] - vtx[(size_t)s * 3 + 2];

  const float q = t0 * (g[0] * t0 + g[1] * t1 + g[2] * t2)
                + t1 * (g[3] * t0 + g[4] * t1 + g[5] * t2)
                + t2 * (g[6] * t0 + g[7] * t1 + g[8] * t2);
  const float w = expf(-q);
  wE[e] = w;
  atomicAdd(&rowsum[s], w);
}

// ---------------------------------------------------------------------------
// Kernel 4: aggregation. One wave per edge; lane handles 4 features (float4
// gather of XW[dst], 4x global_atomic_add_f32 into out[src]).
// ---------------------------------------------------------------------------
__global__ __launch_bounds__(256) void agg_kernel(
    const int* __restrict__ edges,
    const float* __restrict__ wE,
    const float* __restrict__ XW,
    float* __restrict__ out) {
  const int e = blockIdx.x * 8 + (threadIdx.x >> 5);
  if (e >= NE) return;
  const int lane = threadIdx.x & 31;
  const int s = edges[e];
  const int d = edges[NE + e];
  const float w = wE[e];

  const float4 x = *(const float4*)(XW + (size_t)d * NFO + lane * 4);
  float* o = out + (size_t)s * NFO + lane * 4;
  atomicAdd(o + 0, w * x.x);
  atomicAdd(o + 1, w * x.y);
  atomicAdd(o + 2, w * x.z);
  atomicAdd(o + 3, w * x.w);
}

// ---------------------------------------------------------------------------
// Kernel 5: out = agg / (rowsum + eps) + bias   (in-place on d_out)
// ---------------------------------------------------------------------------
__global__ __launch_bounds__(256) void finalize_kernel(
    const float* __restrict__ rowsum,
    const float* __restrict__ bias,
    float* __restrict__ out) {
  const int i = blockIdx.x * blockDim.x + threadIdx.x;
  if (i >= NV * NFO) return;
  const int row = i >> 7;           // NFO == 128
  const int col = i & (NFO - 1);
  out[i] = out[i] / (rowsum[row] + EPSV) + bias[col];
}

// ---------------------------------------------------------------------------
extern "C" void kernel_launch(void* const* d_in, const int* in_sizes, int n_in,
                              void* d_out, int out_size, void* d_ws, size_t ws_size,
                              hipStream_t stream) {
  const float* feat  = (const float*)d_in[0];
  const float* vtx   = (const float*)d_in[1];
  const int*   edges = (const int*)d_in[2];
  // d_in[3] = faces, unused by the reference math
  const float* W1    = (const float*)d_in[4];
  const float* b1    = (const float*)d_in[5];
  const float* W2    = (const float*)d_in[6];
  const float* b2    = (const float*)d_in[7];
  const float* Wc    = (const float*)d_in[8];
  const float* bias  = (const float*)d_in[9];

  float* out = (float*)d_out;
  float* ws  = (float*)d_ws;
  float* XW     = ws;                                // NV*NFO = 6,400,000 f32
  float* G      = XW + (size_t)NV * NFO;             // NV*9   =   450,000 f32
  float* wE     = G + (size_t)NV * 9;                // NE     =   800,000 f32
  float* rowsum = wE + (size_t)NE;                   // NV     =    50,000 f32

  hipMemsetAsync(out, 0, (size_t)NV * NFO * sizeof(float), stream);
  hipMemsetAsync(rowsum, 0, (size_t)NV * sizeof(float), stream);

  metric_kernel<<<(NV + MBLK - 1) / MBLK, MBLK, 0, stream>>>(feat, W1, b1, W2, b2, G);
  xw_wmma<<<NV / 16, 256, 0, stream>>>(feat, Wc, XW);
  edge_kernel<<<(NE + 255) / 256, 256, 0, stream>>>(vtx, edges, G, wE, rowsum);
  agg_kernel<<<NE / 8, 256, 0, stream>>>(edges, wE, XW, out);
  finalize_kernel<<<(NV * NFO + 255) / 256, 256, 0, stream>>>(rowsum, bias, out);
}